// InvariantMessage_52553219834200
// MI455X (gfx1250) — compile-verified
//
#include <hip/hip_runtime.h>
#include <math.h>

typedef __attribute__((ext_vector_type(2))) float v2f;
typedef __attribute__((ext_vector_type(8))) float v8f;

#define FEAT   128
#define NOUT   384     // 3*FEAT
#define NRBF   20
#define PI_F   3.14159265358979323846f
#define RCUT   5.0f
#define LDS_STRIDE 132 // FEAT+4: conflict-free LDS column reads
#define EDGE_GROUPS 8  // 8 x 16 = 128 edges per block in kernel 2

__device__ __forceinline__ v8f wmma_f32_16x16x4(v2f a, v2f b, v8f c) {
    // D = A(16x4) * B(4x16) + C(16x16), all f32, wave32
    return __builtin_amdgcn_wmma_f32_16x16x4_f32(
        /*neg_a=*/false, a, /*neg_b=*/false, b,
        /*c_mod=*/(short)0, c, /*reuse_a=*/false, /*reuse_b=*/false);
}

// ---------------------------------------------------------------------------
// Kernel 1: h = silu(s_j @ W1 + b1) @ W2 + b2        (h -> d_ws, 76.8 MB, L2)
// One block = 16 nodes, 256 threads = 8 waves.
// ---------------------------------------------------------------------------
__global__ __launch_bounds__(256) void node_mlp_kernel(
    const float* __restrict__ s_j, const float* __restrict__ W1,
    const float* __restrict__ b1,  const float* __restrict__ W2,
    const float* __restrict__ b2,  float* __restrict__ h, int n_nodes) {

    __shared__ float s_tile[16 * LDS_STRIDE]; // input tile 16x128
    __shared__ float t_tile[16 * LDS_STRIDE]; // silu(x@W1+b1) tile 16x128

    const int tid  = threadIdx.x;
    const int lane = tid & 31;
    const int wave = tid >> 5;
    const int col  = lane & 15;   // N column within tile / A row
    const int hi   = lane >> 4;   // half-wave select
    const int node0 = blockIdx.x * 16;

    // Cooperative coalesced load of 16x128 input rows into LDS.
    for (int i = 0; i < 8; ++i) {
        int off = i * 256 + tid;          // 0..2047
        int r = off >> 7, cc = off & 127; // row, col
        s_tile[r * LDS_STRIDE + cc] = s_j[(size_t)(node0 + r) * FEAT + cc];
    }
    __syncthreads();

    // ---- GEMM1 + silu: each wave owns one 16x16 tile of the 16x128 result.
    {
        const int n0 = wave * 16;
        const float bn = b1[n0 + col];
        v8f c = {bn, bn, bn, bn, bn, bn, bn, bn};
        #pragma unroll 4
        for (int k0 = 0; k0 < FEAT; k0 += 4) {
            const int ka = k0 + 2 * hi;
            v2f a, b;
            a.x = s_tile[col * LDS_STRIDE + ka + 0];
            a.y = s_tile[col * LDS_STRIDE + ka + 1];
            b.x = W1[(ka + 0) * FEAT + n0 + col];
            b.y = W1[(ka + 1) * FEAT + n0 + col];
            c = wmma_f32_16x16x4(a, b, c);
        }
        #pragma unroll
        for (int v = 0; v < 8; ++v) {
            float x = c[v];
            float s = x / (1.0f + expf(-x)); // silu
            t_tile[(v + 8 * hi) * LDS_STRIDE + n0 + col] = s;
        }
    }
    __syncthreads();

    // ---- GEMM2: 24 N-tiles over N=384, 3 per wave.
    for (int t = 0; t < 3; ++t) {
        const int n0 = (wave * 3 + t) * 16;
        const float bn = b2[n0 + col];
        v8f c = {bn, bn, bn, bn, bn, bn, bn, bn};
        #pragma unroll 4
        for (int k0 = 0; k0 < FEAT; k0 += 4) {
            const int ka = k0 + 2 * hi;
            v2f a, b;
            a.x = t_tile[col * LDS_STRIDE + ka + 0];
            a.y = t_tile[col * LDS_STRIDE + ka + 1];
            b.x = W2[(ka + 0) * NOUT + n0 + col];
            b.y = W2[(ka + 1) * NOUT + n0 + col];
            c = wmma_f32_16x16x4(a, b, c);
        }
        #pragma unroll
        for (int v = 0; v < 8; ++v) {
            int m = node0 + v + 8 * hi;
            if (m < n_nodes) h[(size_t)m * NOUT + n0 + col] = c[v];
        }
    }
}

// ---------------------------------------------------------------------------
// Kernel 2: out[e] = h[nbrs[e,1]] * ((rbf(e) @ Wd + bd) * env(e))
// One block = 128 edges (8 groups of 16), 256 threads = 8 waves.
// Each wave owns 3 N-tiles; its Wd B-fragments (30 VGPRs) + bd bias are
// loaded ONCE per block and reused across all 8 edge groups -> 8x less
// weight traffic, keeping L2 bandwidth free for the phi gather.
// Output streamed with nontemporal stores so h stays resident in L2.
// ---------------------------------------------------------------------------
__global__ __launch_bounds__(256) void edge_kernel(
    const float* __restrict__ dist, const int* __restrict__ nbrs,
    const float* __restrict__ Wd,   const float* __restrict__ bd,
    const float* __restrict__ h,    float* __restrict__ out, int n_edges) {

    const int tid  = threadIdx.x;
    const int lane = tid & 31;
    const int wave = tid >> 5;
    const int col  = lane & 15;
    const int hi   = lane >> 4;

    // ---- Hoisted per-wave B fragments + bias for this wave's 3 N-tiles.
    v2f   bfrag[3][5];
    float bias[3];
    int   ncol[3];
    #pragma unroll
    for (int t = 0; t < 3; ++t) {
        const int n = (wave * 3 + t) * 16 + col;
        ncol[t] = n;
        bias[t] = bd[n];
        #pragma unroll
        for (int kk = 0; kk < 5; ++kk) {
            const int kb = kk * 4 + 2 * hi;       // k index 0..19
            bfrag[t][kk].x = Wd[(kb + 0) * NOUT + n];
            bfrag[t][kk].y = Wd[(kb + 1) * NOUT + n];
        }
    }

    const int eblk = blockIdx.x * (16 * EDGE_GROUPS);

    for (int g = 0; g < EDGE_GROUPS; ++g) {
        const int e0 = eblk + g * 16;

        // ---- A fragment: rbf row for edge (e0+col), K striped per f32 layout.
        int ea = e0 + col; if (ea >= n_edges) ea = n_edges - 1;
        const float da = dist[ea];
        const float inv_d = 1.0f / da;
        const float wphase = PI_F * da / RCUT;
        v2f afrag[5];
        #pragma unroll
        for (int kk = 0; kk < 5; ++kk) {
            const int kb = kk * 4 + 2 * hi;
            afrag[kk].x = sinf((float)(kb + 1) * wphase) * inv_d;
            afrag[kk].y = sinf((float)(kb + 2) * wphase) * inv_d;
        }

        // ---- Per-output-row envelope + gather index.
        float envv[8];
        int   jrow[8];
        #pragma unroll
        for (int v = 0; v < 8; ++v) {
            int e = e0 + v + 8 * hi; if (e >= n_edges) e = n_edges - 1;
            const float dm = dist[e];
            envv[v] = (dm < RCUT) ? 0.5f * (cosf(PI_F * dm / RCUT) + 1.0f) : 0.0f;
            jrow[v] = nbrs[2 * e + 1];
        }

        // ---- Accumulate: kk-outer / tile-inner interleaves the 3 chains
        // (same A operand back-to-back, dependent WMMAs spaced by 3).
        v8f c[3];
        #pragma unroll
        for (int t = 0; t < 3; ++t) {
            const float bn = bias[t];
            c[t] = (v8f){bn, bn, bn, bn, bn, bn, bn, bn};
        }
        #pragma unroll
        for (int kk = 0; kk < 5; ++kk) {
            #pragma unroll
            for (int t = 0; t < 3; ++t) {
                c[t] = wmma_f32_16x16x4(afrag[kk], bfrag[t][kk], c[t]);
            }
        }

        // ---- Gather phi from L2-resident h, scale, stream out.
        #pragma unroll
        for (int t = 0; t < 3; ++t) {
            const int n = ncol[t];
            #pragma unroll
            for (int v = 0; v < 8; ++v) {
                const int e = e0 + v + 8 * hi;
                if (e < n_edges) {
                    const float phi = h[(size_t)jrow[v] * NOUT + n]; // L2 gather
                    const float val = c[t][v] * envv[v] * phi;
                    __builtin_nontemporal_store(val, &out[(size_t)e * NOUT + n]);
                }
            }
        }
    }
}

extern "C" void kernel_launch(void* const* d_in, const int* in_sizes, int n_in,
                              void* d_out, int out_size, void* d_ws, size_t ws_size,
                              hipStream_t stream) {
    const float* s_j  = (const float*)d_in[0];
    const float* dist = (const float*)d_in[1];
    const int*   nbrs = (const int*)d_in[2];
    const float* W1   = (const float*)d_in[3];
    const float* b1   = (const float*)d_in[4];
    const float* W2   = (const float*)d_in[5];
    const float* b2   = (const float*)d_in[6];
    const float* Wd   = (const float*)d_in[7];
    const float* bd   = (const float*)d_in[8];

    const int n_nodes = in_sizes[0] / FEAT;  // 50000
    const int n_edges = in_sizes[1];         // 800000

    float* h = (float*)d_ws;                 // 50000*384*4 = 76.8 MB scratch

    const int nblk = (n_nodes + 15) / 16;
    const int eblk = (n_edges + 16 * EDGE_GROUPS - 1) / (16 * EDGE_GROUPS);
    node_mlp_kernel<<<nblk, 256, 0, stream>>>(s_j, W1, b1, W2, b2, h, n_nodes);
    edge_kernel<<<eblk, 256, 0, stream>>>(dist, nbrs, Wd, bd, h, (float*)d_out, n_edges);
}